// SuperLoss_12953621365466
// MI455X (gfx1250) — compile-verified
//
#include <hip/hip_runtime.h>
#include <cmath>

typedef __attribute__((ext_vector_type(2))) float v2f;
typedef __attribute__((ext_vector_type(8))) float v8f;

// Faithful port of the reference lambertw0 (branch-point series / log1p init,
// 30 Halley iterations with the same guards).
__device__ __forceinline__ float lambertw0_halley(float y, float ef) {
  float p  = sqrtf(fmaxf(2.0f * (1.0f + ef * y), 0.0f));
  float wn = -1.0f + p - (p * p) / 3.0f;
  float wf = log1pf(fmaxf(y, 0.0f));
  float w  = (y < 0.0f) ? wn : wf;
#pragma unroll 1
  for (int i = 0; i < 30; ++i) {
    float ew    = expf(w);
    float f     = w * ew - y;
    float wp1   = w + 1.0f;
    float swp1  = (wp1 == 0.0f) ? 1.0f : wp1;
    float denom = ew * wp1 - (w + 2.0f) * f / (2.0f * swp1);
    float sden  = (denom == 0.0f) ? 1.0f : denom;
    float wnew  = w - f / sden;
    w = (fabsf(f) < 1e-30f) ? w : wnew;
  }
  return w;
}

// Kernel 1: elementwise SuperLoss with per-wave compaction of the expensive
// (unclamped) Lambert-W evaluations; one fp32 partial sum per block.
__global__ __launch_bounds__(256) void superloss_main(
    const float* __restrict__ li, float* __restrict__ ws,
    int n, int iters, float tau, float xc, float ef) {
  __shared__ float q[8][64];     // per-wave compaction queue of y values
  __shared__ float wsum[8];

  const int lane = threadIdx.x & 31;
  const int wave = threadIdx.x >> 5;
  const unsigned stride = gridDim.x * blockDim.x;
  unsigned idx = blockIdx.x * blockDim.x + threadIdx.x;

  // Clamped-branch constants: y = 0.5*(-2/e) is shared by every clamped
  // element, so run the exact 30-iteration pipeline once (wave-uniform).
  const float yc   = 0.5f * xc;
  const float wcl  = lambertw0_halley(yc, ef);
  const float sigc = expf(-wcl);
  const float lsc  = logf(sigc);
  const float tc   = 0.25f * lsc * lsc;   // LAM * log(sigma_c)^2

  float acc = 0.0f;
  int qn = 0;                              // queue depth (wave-uniform)

  for (int k = 0; k < iters; ++k, idx += stride) {
    const bool  valid = idx < (unsigned)n;
    const float l     = valid ? li[idx] : 0.0f;
    const float d     = l - tau;           // l - TAU
    const float beta  = d * 4.0f;          // (l - TAU)/LAM, exact
    const bool  clamped = valid && (beta < xc);
    const bool  uncl    = valid && !(beta < xc);  // NaN routes to queue

    if (clamped) acc += d * sigc + tc;

    const unsigned mask = (unsigned)__ballot(uncl);
    if (uncl) {
      int pos = qn + __popc(mask & ((1u << lane) - 1u));
      q[wave][pos] = 2.0f * d;             // y = 0.5*beta, exact
    }
    qn += __popc(mask);

    if (qn >= 32) {                        // wave-uniform: full batch ready
      qn -= 32;
      const float y   = q[wave][qn + lane];
      const float w   = lambertw0_halley(y, ef);
      const float sig = expf(-w);
      const float ls  = logf(sig);
      acc += (0.5f * y) * sig + 0.25f * ls * ls;   // (l-TAU)==y/2 exactly
    }
  }

  if (qn > 0) {                            // drain partial batch
    const float y   = (lane < qn) ? q[wave][lane] : 1.0f;
    const float w   = lambertw0_halley(y, ef);
    const float sig = expf(-w);
    const float ls  = logf(sig);
    const float c   = (0.5f * y) * sig + 0.25f * ls * ls;
    if (lane < qn) acc += c;
  }

  for (int off = 16; off > 0; off >>= 1) acc += __shfl_xor(acc, off, 32);
  if (lane == 0) wsum[wave] = acc;
  __syncthreads();
  if (threadIdx.x == 0) {
    float s = 0.0f;
    for (int i = 0; i < 8; ++i) s += wsum[i];
    ws[blockIdx.x] = s;
  }
}

// Kernel 2: single wave32 finisher. Sum P block partials, then reduce via
// V_WMMA_F32_16X16X4_F32 with B=ones, C=0: sum(all of D) == 16*sum(all of A),
// which is independent of the exact lane/VGPR operand layout.
__global__ __launch_bounds__(32) void superloss_finish(
    const float* __restrict__ ws, float* __restrict__ out,
    int p, float inv_scale) {
  const int lane = threadIdx.x;            // one full wave32, EXEC all ones
  float a0 = 0.0f, a1 = 0.0f;
  const int iters = (p + 63) / 64;
  for (int k = 0; k < iters; ++k) {        // uniform trip count (reconverges)
    const int i0 = k * 64 + lane;
    const int i1 = i0 + 32;
    if (i0 < p) a0 += ws[i0];
    if (i1 < p) a1 += ws[i1];
  }
  v2f A; A.x = a0;   A.y = a1;
  v2f B; B.x = 1.0f; B.y = 1.0f;
  v8f C = {};
  v8f D = __builtin_amdgcn_wmma_f32_16x16x4_f32(
      /*neg_a=*/false, A, /*neg_b=*/false, B,
      /*c_mod=*/(short)0, C, /*reuse_a=*/false, /*reuse_b=*/false);
  float s = D[0] + D[1] + D[2] + D[3] + D[4] + D[5] + D[6] + D[7];
  for (int off = 16; off > 0; off >>= 1) s += __shfl_xor(s, off, 32);
  if (lane == 0) out[0] = s * inv_scale;   // inv_scale = 1/(16*N)
}

extern "C" void kernel_launch(void* const* d_in, const int* in_sizes, int n_in,
                              void* d_out, int out_size, void* d_ws, size_t ws_size,
                              hipStream_t stream) {
  const float* li  = (const float*)d_in[0];
  float*       out = (float*)d_out;
  float*       ws  = (float*)d_ws;
  const int n = in_sizes[0];

  int blocks = 2048;
  const size_t maxb = ws_size / sizeof(float);
  if ((size_t)blocks > maxb) blocks = (int)maxb;
  if (blocks < 1) blocks = 1;
  const int tpb = 256;
  const long stride = (long)blocks * tpb;
  const int iters = (int)(((long)n + stride - 1) / stride);

  // Match jnp.float32 constants exactly: round host doubles to fp32.
  const double E_D = 2.718281828459045;
  const float tau = (float)log(330.0);     // float(np.log(330.0))
  const float xc  = (float)(-2.0 / E_D);   // jnp.float32(-2/e)
  const float ef  = (float)E_D;            // jnp.float32(np.e)

  superloss_main<<<blocks, tpb, 0, stream>>>(li, ws, n, iters, tau, xc, ef);

  const float inv_scale = 1.0f / (16.0f * (float)n);  // exact power-of-2 scale
  superloss_finish<<<1, 32, 0, stream>>>(ws, out, blocks, inv_scale);
}